// InvariantPointAttention_80109730005209
// MI455X (gfx1250) — compile-verified
//
#include <hip/hip_runtime.h>
#include <hip/hip_bf16.h>
#include <math.h>
#include <stdint.h>

// Problem constants (from reference)
#define L_  1024
#define D_  384
#define H_  4
#define C_  128
#define PQK_ 4
#define PV_  8

typedef __attribute__((ext_vector_type(2))) float v2f;
typedef __attribute__((ext_vector_type(8))) float v8f;

// ---------------------------------------------------------------------------
// One wave computes a 16x64 f32 tile (4 accumulators) of A[MxK] @ B[KxN] with
// V_WMMA_F32_16X16X4_F32, K stepped by 4. The A fragment is shared by the 4
// column-tiles (4x reuse per A load).
// f32 A-fragment layout (ISA 7.12.2): lanes 0-15 -> rows, VGPR0 holds K=2*half,
// VGPR1 holds K=2*half+1 (half = lane>=16); B mirrors with N across lanes.
// Columns >= ncolsB are CLAMPED (not masked): garbage values land only in C
// columns that are never stored, so the inner loop is branch-free and EXEC
// stays all-ones (required for WMMA anyway).
// ---------------------------------------------------------------------------
template <bool BT>
__device__ __forceinline__ void wmma_tile64(const float* __restrict__ A, long lda,
                                            const float* __restrict__ B, long ldb,
                                            int K, int row0, int col0, int ncolsB,
                                            v8f acc[4]) {
  const int lane = threadIdx.x & 31;
  const int r = lane & 15;
  const int half = lane >> 4;
  const float* arow = A + (long)(row0 + r) * lda;
  long boff[4];
#pragma unroll
  for (int j = 0; j < 4; ++j) {
    int bc = col0 + 16 * j + r;
    bc = (bc < ncolsB) ? bc : (ncolsB - 1);  // clamp: safe reads, unstored cols
    boff[j] = BT ? (long)bc * ldb : (long)bc;
    acc[j] = (v8f){};
  }
  for (int k0 = 0; k0 < K; k0 += 4) {
    const int ka = k0 + 2 * half;
    v2f a;
    a.x = arow[ka];
    a.y = arow[ka + 1];
#pragma unroll
    for (int j = 0; j < 4; ++j) {
      v2f b;
      if (BT) {
        b.x = B[boff[j] + ka];
        b.y = B[boff[j] + ka + 1];
      } else {
        b.x = B[(long)ka * ldb + boff[j]];
        b.y = B[(long)(ka + 1) * ldb + boff[j]];
      }
      acc[j] = __builtin_amdgcn_wmma_f32_16x16x4_f32(false, a, false, b, (short)0,
                                                     acc[j], false, false);
    }
  }
}

__device__ __forceinline__ void store_tile64(float* __restrict__ Cp, long ldc,
                                             int row0, int col0, int ncols,
                                             const v8f acc[4]) {
  const int lane = threadIdx.x & 31;
  const int r = lane & 15;
  const int half = lane >> 4;
#pragma unroll
  for (int j = 0; j < 4; ++j) {
    const int c = col0 + 16 * j + r;
    if (c < ncols) {
#pragma unroll
      for (int i = 0; i < 8; ++i)
        Cp[(long)(row0 + i + 8 * half) * ldc + c] = acc[j][i];
    }
  }
}

// ---------------------------------------------------------------------------
// C[M,N] = A[M,K] @ W[K,N] + b  (projections). grid = (ceil(N/64), M/16), blk 32.
// ---------------------------------------------------------------------------
__global__ void __launch_bounds__(32) gemm_bias(const float* __restrict__ A,
                                                const float* __restrict__ W,
                                                const float* __restrict__ bias,
                                                float* __restrict__ Cmat,
                                                int N, int K) {
  const int col0 = blockIdx.x * 64, row0 = blockIdx.y * 16;
  v8f acc[4];
  wmma_tile64<false>(A, K, W, N, K, row0, col0, N, acc);
  const int lane = threadIdx.x & 31, r = lane & 15, half = lane >> 4;
#pragma unroll
  for (int j = 0; j < 4; ++j) {
    const int c = col0 + 16 * j + r;
    if (c < N) {
      const float bv = bias[c];
#pragma unroll
      for (int i = 0; i < 8; ++i)
        Cmat[(long)(row0 + i + 8 * half) * N + c] = acc[j][i] + bv;
    }
  }
}

// ---------------------------------------------------------------------------
// Rotate/translate points to global frame; accumulate qn/kn norms.
// grid = L, block = 32. qp/kp: [n][48]=[h][4][3]; vp: [n][96]=[h][8][3]
// ---------------------------------------------------------------------------
__global__ void frames_kernel(const float* __restrict__ qp, const float* __restrict__ kp,
                              const float* __restrict__ vp, const float* __restrict__ R,
                              const float* __restrict__ t, float* __restrict__ qpg,
                              float* __restrict__ kpg, float* __restrict__ vpg,
                              float* __restrict__ qn, float* __restrict__ kn) {
  __shared__ float sq[48], sk[48];
  const int n = blockIdx.x;
  const int lane = threadIdx.x;
  float Rm[9], tv[3];
#pragma unroll
  for (int i = 0; i < 9; ++i) Rm[i] = R[n * 9 + i];
#pragma unroll
  for (int i = 0; i < 3; ++i) tv[i] = t[n * 3 + i];

  {  // qp (lanes 0-15) / kp (lanes 16-31): 16 points each
    const int p = (lane < 16) ? lane : lane - 16;
    const float* src = ((lane < 16) ? qp : kp) + n * 48 + p * 3;
    const float x = src[0], y = src[1], z = src[2];
    const float gx = Rm[0] * x + Rm[1] * y + Rm[2] * z + tv[0];
    const float gy = Rm[3] * x + Rm[4] * y + Rm[5] * z + tv[1];
    const float gz = Rm[6] * x + Rm[7] * y + Rm[8] * z + tv[2];
    float* dst = ((lane < 16) ? qpg : kpg) + n * 48 + p * 3;
    dst[0] = gx; dst[1] = gy; dst[2] = gz;
    float* s = ((lane < 16) ? sq : sk) + p * 3;
    s[0] = gx; s[1] = gy; s[2] = gz;
  }
  {  // vp: 32 points, one per lane
    const float* src = vp + n * 96 + lane * 3;
    const float x = src[0], y = src[1], z = src[2];
    float* dst = vpg + n * 96 + lane * 3;
    dst[0] = Rm[0] * x + Rm[1] * y + Rm[2] * z + tv[0];
    dst[1] = Rm[3] * x + Rm[4] * y + Rm[5] * z + tv[1];
    dst[2] = Rm[6] * x + Rm[7] * y + Rm[8] * z + tv[2];
  }
  __syncthreads();
  if (lane < H_) {
    float sa = 0.0f, sb = 0.0f;
#pragma unroll
    for (int j = 0; j < 12; ++j) {
      const float a = sq[lane * 12 + j], b = sk[lane * 12 + j];
      sa += a * a; sb += b * b;
    }
    qn[n * H_ + lane] = sa;
    kn[n * H_ + lane] = sb;
  }
}

// ---------------------------------------------------------------------------
// pair_bias: logits[h][n][m] = pair[n,m,:] @ Wpb[:,h] + bpb[h]
// Bandwidth-dominant kernel: 512 MB stream.  Double-buffered async pipeline:
//   global_load_async_to_lds_b128 prefetches stage s+1 (32 KB) into the
//   alternate LDS buffer while stage s is reduced from LDS.
// grid = 2048 blocks x 256 thr (8 waves); 512 pairs/block, 64 pairs/stage.
// ---------------------------------------------------------------------------
#define PB_PAIRS_PER_BLOCK 512
#define PB_STAGE_PAIRS 64
#define PB_STAGES (PB_PAIRS_PER_BLOCK / PB_STAGE_PAIRS)  // 8

__global__ void __launch_bounds__(256) pair_bias_kernel(const float* __restrict__ pair,
                                                        const float* __restrict__ Wpb,
                                                        const float* __restrict__ bpb,
                                                        float* __restrict__ logits) {
  __shared__ float lds[2][PB_STAGE_PAIRS * 128];  // 2 x 32 KB
  const int tid = threadIdx.x;
  const int lane = tid & 31;
  const int wv = tid >> 5;
  const long pairBase = (long)blockIdx.x * PB_PAIRS_PER_BLOCK;

  float w[4][4];
#pragma unroll
  for (int j = 0; j < 4; ++j)
#pragma unroll
    for (int h = 0; h < 4; ++h) w[j][h] = Wpb[(lane * 4 + j) * 4 + h];
  const float bb = (lane < 4) ? bpb[lane] : 0.0f;

  const unsigned lbase = (unsigned)(uintptr_t)(void*)&lds[0][0];

  // Issue one stage: 32 KB block copy, 8 async b128 per thread (8 per wave).
  auto issue_stage = [&](int s, int buf) {
    const float* gsrc = pair + (pairBase + (long)s * PB_STAGE_PAIRS) * 128 + tid * 4;
    const unsigned loff = lbase + (unsigned)buf * (PB_STAGE_PAIRS * 128 * 4) + tid * 16;
#pragma unroll
    for (int rr = 0; rr < 8; ++rr) {
      asm volatile("global_load_async_to_lds_b128 %0, %1, off"
                   :: "v"(loff + rr * 4096), "v"(gsrc + rr * 1024)
                   : "memory");
    }
  };

  issue_stage(0, 0);
  for (int s = 0; s < PB_STAGES; ++s) {
    const int buf = s & 1;
    if (s + 1 < PB_STAGES) {
      issue_stage(s + 1, buf ^ 1);
      // stage s's 8 async ops done; stage s+1's 8 may remain in flight
      asm volatile("s_wait_asynccnt 0x8" ::: "memory");
    } else {
      asm volatile("s_wait_asynccnt 0x0" ::: "memory");
    }
    __syncthreads();  // all waves' async data for stage s visible

    // Wave wv reduces local pairs [wv*8, wv*8+8): 32 lanes x float4 = 128 c's.
#pragma unroll
    for (int q = 0; q < 8; ++q) {
      const int pl = wv * 8 + q;
      const float4 v = *(const float4*)&lds[buf][pl * 128 + lane * 4];
      float s0 = v.x * w[0][0] + v.y * w[1][0] + v.z * w[2][0] + v.w * w[3][0];
      float s1 = v.x * w[0][1] + v.y * w[1][1] + v.z * w[2][1] + v.w * w[3][1];
      float s2 = v.x * w[0][2] + v.y * w[1][2] + v.z * w[2][2] + v.w * w[3][2];
      float s3 = v.x * w[0][3] + v.y * w[1][3] + v.z * w[2][3] + v.w * w[3][3];
#pragma unroll
      for (int off = 16; off > 0; off >>= 1) {
        s0 += __shfl_xor(s0, off, 32);
        s1 += __shfl_xor(s1, off, 32);
        s2 += __shfl_xor(s2, off, 32);
        s3 += __shfl_xor(s3, off, 32);
      }
      if (lane < 4) {
        const float sv = (lane == 0) ? s0 : (lane == 1) ? s1 : (lane == 2) ? s2 : s3;
        logits[((long)lane << 20) + pairBase + (long)s * PB_STAGE_PAIRS + pl] = sv + bb;
      }
    }
    __syncthreads();  // buffer reuse fence before stage s+2 overwrites buf
  }
}

// ---------------------------------------------------------------------------
// logits[h][n][m] = (QK/sqrt(C) + cross - 0.5(qn+kn) + pair_bias)/sqrt(H)
// QK: WMMA K=128 (B transposed from ks); cross: WMMA K=12 on global points.
// 16x64 tile over m.  grid = (16, 64, 4), block 32.
// ---------------------------------------------------------------------------
__global__ void __launch_bounds__(32) logits_kernel(
    const float* __restrict__ qs, const float* __restrict__ ks,
    const float* __restrict__ qpg, const float* __restrict__ kpg,
    const float* __restrict__ qn, const float* __restrict__ kn,
    float* __restrict__ logits) {
  const int h = blockIdx.z;
  const int m0 = blockIdx.x * 64, n0 = blockIdx.y * 16;
  v8f qk[4], cr[4];
  wmma_tile64<true>(qs + h * C_, H_ * C_, ks + h * C_, H_ * C_, C_, n0, m0, L_, qk);
  wmma_tile64<true>(qpg + h * 12, 48, kpg + h * 12, 48, 12, n0, m0, L_, cr);
  const int lane = threadIdx.x & 31, r = lane & 15, half = lane >> 4;
  float knm[4];
#pragma unroll
  for (int j = 0; j < 4; ++j) knm[j] = kn[(m0 + 16 * j + r) * H_ + h];
  float* lp = logits + ((long)h << 20);
  const float inv_sqrt_c = 0.088388347648318447f;  // 1/sqrt(128)
#pragma unroll
  for (int i = 0; i < 8; ++i) {
    const int n = n0 + i + 8 * half;
    const float qnn = qn[n * H_ + h];
    float* lrow = lp + ((long)n << 10);
#pragma unroll
    for (int j = 0; j < 4; ++j) {
      const int m = m0 + 16 * j + r;
      const float pb = lrow[m];  // written by pair_bias_kernel
      lrow[m] = (qk[j][i] * inv_sqrt_c + cr[j][i] - 0.5f * (qnn + knm[j]) + pb) * 0.5f;
    }
  }
}

// ---------------------------------------------------------------------------
// Row softmax over m (1024). grid = H*L blocks x 256 threads.
// ---------------------------------------------------------------------------
__global__ void softmax_kernel(float* __restrict__ logits) {
  __shared__ float red[8];
  float* p = logits + ((long)blockIdx.x << 10);
  const int tid = threadIdx.x;
  float x0 = p[tid], x1 = p[tid + 256], x2 = p[tid + 512], x3 = p[tid + 768];
  float mx = fmaxf(fmaxf(x0, x1), fmaxf(x2, x3));
#pragma unroll
  for (int o = 16; o > 0; o >>= 1) mx = fmaxf(mx, __shfl_xor(mx, o, 32));
  if ((tid & 31) == 0) red[tid >> 5] = mx;
  __syncthreads();
  float gmx = red[0];
#pragma unroll
  for (int i = 1; i < 8; ++i) gmx = fmaxf(gmx, red[i]);
  const float e0 = __expf(x0 - gmx), e1 = __expf(x1 - gmx);
  const float e2 = __expf(x2 - gmx), e3 = __expf(x3 - gmx);
  float s = e0 + e1 + e2 + e3;
#pragma unroll
  for (int o = 16; o > 0; o >>= 1) s += __shfl_xor(s, o, 32);
  __syncthreads();
  if ((tid & 31) == 0) red[tid >> 5] = s;
  __syncthreads();
  float tot = 0.0f;
#pragma unroll
  for (int i = 0; i < 8; ++i) tot += red[i];
  const float inv = 1.0f / tot;
  p[tid] = e0 * inv; p[tid + 256] = e1 * inv;
  p[tid + 512] = e2 * inv; p[tid + 768] = e3 * inv;
}

// ---------------------------------------------------------------------------
// attn[h] @ V  (K=1024). Per-head B/dst offsets passed in.
// grid = (ceil(ncols/64), 64, 4), block 32.
// ---------------------------------------------------------------------------
__global__ void __launch_bounds__(32) attn_gemm(const float* __restrict__ attn,
                                                const float* __restrict__ Bm,
                                                float* __restrict__ dst,
                                                int ldb, int b_off, int ncols,
                                                int ldc, int d_off) {
  const int h = blockIdx.z;
  const float* A = attn + ((long)h << 20);
  const int col0 = blockIdx.x * 64, row0 = blockIdx.y * 16;
  v8f acc[4];
  wmma_tile64<false>(A, L_, Bm + h * b_off, ldb, L_, row0, col0, ncols, acc);
  store_tile64(dst + h * d_off, ldc, row0, col0, ncols, acc);
}

// ---------------------------------------------------------------------------
// Back to local frame: R^T @ (x - t); write into out_cat[:, 512 + ...].
// One thread per (n, h*8+a) = 32768 threads.
// ---------------------------------------------------------------------------
__global__ void local_frame_kernel(const float* __restrict__ opoint,
                                   const float* __restrict__ R,
                                   const float* __restrict__ t,
                                   float* __restrict__ out_cat) {
  const int idx = blockIdx.x * blockDim.x + threadIdx.x;
  const int n = idx >> 5;
  const int ha = idx & 31;
  const float dx = opoint[n * 96 + ha * 3 + 0] - t[n * 3 + 0];
  const float dy = opoint[n * 96 + ha * 3 + 1] - t[n * 3 + 1];
  const float dz = opoint[n * 96 + ha * 3 + 2] - t[n * 3 + 2];
  const float* Rn = R + n * 9;
  float* o = out_cat + n * 608 + 512 + ha * 3;
  o[0] = Rn[0] * dx + Rn[3] * dy + Rn[6] * dz;
  o[1] = Rn[1] * dx + Rn[4] * dy + Rn[7] * dz;
  o[2] = Rn[2] * dx + Rn[5] * dy + Rn[8] * dz;
}

// ---------------------------------------------------------------------------
// out = single + out_cat @ Wo + bo.  grid = (6, 64), block 32.
// ---------------------------------------------------------------------------
__global__ void __launch_bounds__(32) final_gemm(const float* __restrict__ cat,
                                                 const float* __restrict__ Wo,
                                                 const float* __restrict__ bo,
                                                 const float* __restrict__ single,
                                                 float* __restrict__ out) {
  const int col0 = blockIdx.x * 64, row0 = blockIdx.y * 16;
  v8f acc[4];
  wmma_tile64<false>(cat, 608, Wo, D_, 608, row0, col0, D_, acc);
  const int lane = threadIdx.x & 31, r = lane & 15, half = lane >> 4;
#pragma unroll
  for (int j = 0; j < 4; ++j) {
    const int c = col0 + 16 * j + r;
    if (c < D_) {
      const float bv = bo[c];
#pragma unroll
      for (int i = 0; i < 8; ++i) {
        const long o = (long)(row0 + i + 8 * half) * D_ + c;
        out[o] = single[o] + acc[j][i] + bv;
      }
    }
  }
}

// ---------------------------------------------------------------------------
extern "C" void kernel_launch(void* const* d_in, const int* in_sizes, int n_in,
                              void* d_out, int out_size, void* d_ws, size_t ws_size,
                              hipStream_t stream) {
  (void)in_sizes; (void)n_in; (void)out_size; (void)ws_size;
  const float* single = (const float*)d_in[0];
  const float* pair   = (const float*)d_in[1];
  const float* R      = (const float*)d_in[2];
  const float* t      = (const float*)d_in[3];
  const float* Wq_s = (const float*)d_in[4];   const float* bq_s = (const float*)d_in[5];
  const float* Wk_s = (const float*)d_in[6];   const float* bk_s = (const float*)d_in[7];
  const float* Wv_s = (const float*)d_in[8];   const float* bv_s = (const float*)d_in[9];
  const float* Wq_p = (const float*)d_in[10];  const float* bq_p = (const float*)d_in[11];
  const float* Wk_p = (const float*)d_in[12];  const float* bk_p = (const float*)d_in[13];
  const float* Wv_p = (const float*)d_in[14];  const float* bv_p = (const float*)d_in[15];
  const float* Wpb  = (const float*)d_in[16];  const float* bpb  = (const float*)d_in[17];
  const float* Wo   = (const float*)d_in[18];  const float* bo   = (const float*)d_in[19];
  float* out = (float*)d_out;

  // Workspace carve-up (floats); total ~6.89M floats = 27.6 MB.
  float* ws  = (float*)d_ws;
  float* qs  = ws;                   // 1024*512
  float* ksb = qs  + 524288;         // 1024*512
  float* vsb = ksb + 524288;         // 1024*512
  float* qp  = vsb + 524288;         // 1024*48
  float* kp  = qp  + 49152;          // 1024*48
  float* vp  = kp  + 49152;          // 1024*96
  float* qpg = vp  + 98304;          // 1024*48
  float* kpg = qpg + 49152;          // 1024*48
  float* vpg = kpg + 49152;          // 1024*96
  float* qn  = vpg + 98304;          // 1024*4
  float* kn  = qn  + 4096;           // 1024*4
  float* logits = kn + 4096;         // 4*1024*1024
  float* ocat   = logits + 4194304;  // 1024*608
  float* opoint = ocat + 622592;     // 1024*96

  const dim3 b32(32);

  // Projections (WMMA GEMMs, K=384), 16x64 tiles
  gemm_bias<<<dim3(8, 64), b32, 0, stream>>>(single, Wq_s, bq_s, qs,  512, D_);
  gemm_bias<<<dim3(8, 64), b32, 0, stream>>>(single, Wk_s, bk_s, ksb, 512, D_);
  gemm_bias<<<dim3(8, 64), b32, 0, stream>>>(single, Wv_s, bv_s, vsb, 512, D_);
  gemm_bias<<<dim3(1, 64), b32, 0, stream>>>(single, Wq_p, bq_p, qp,  48,  D_);
  gemm_bias<<<dim3(1, 64), b32, 0, stream>>>(single, Wk_p, bk_p, kp,  48,  D_);
  gemm_bias<<<dim3(2, 64), b32, 0, stream>>>(single, Wv_p, bv_p, vp,  96,  D_);

  // Global frames + point norms
  frames_kernel<<<dim3(L_), b32, 0, stream>>>(qp, kp, vp, R, t, qpg, kpg, vpg, qn, kn);

  // Pair bias (512 MB stream, async-to-LDS double-buffered) -> logits buffer
  pair_bias_kernel<<<dim3(2048), dim3(256), 0, stream>>>(pair, Wpb, bpb, logits);

  // Attention logits (WMMA) + softmax
  logits_kernel<<<dim3(16, 64, 4), b32, 0, stream>>>(qs, ksb, qpg, kpg, qn, kn, logits);
  softmax_kernel<<<dim3(4096), dim3(256), 0, stream>>>(logits);

  // attn @ V (WMMA, K=1024): scalar values (N=128) and point values (N=24)
  attn_gemm<<<dim3(2, 64, 4), b32, 0, stream>>>(logits, vsb, ocat,   512, 128, 128, 608, 128);
  attn_gemm<<<dim3(1, 64, 4), b32, 0, stream>>>(logits, vpg, opoint,  96,  24,  24,  96,  24);

  // Rotate points back to local frame, concat, final projection + residual
  local_frame_kernel<<<dim3(128), dim3(256), 0, stream>>>(opoint, R, t, ocat);
  final_gemm<<<dim3(6, 64), b32, 0, stream>>>(ocat, Wo, bo, single, out);
}